// DifferentiableDefocusRenderer_47201690583169
// MI455X (gfx1250) — compile-verified
//
#include <hip/hip_runtime.h>
#include <stdint.h>

// ---------------------------------------------------------------------------
// Defocus renderer for MI455X (gfx1250, wave32).
//
// result(b,c,y,x) = Gaussian_blur_{plane(b,y,x)}(img)(b,c,y,x),
// plane = round(coc / (50/31)) clamped to [0,31], kernels separable (g x g),
// zero 'same' padding, kernel width <= 31 (halo 15).
//
// Per image row, the horizontal blur for 16 planes is one 16x16 =
// (16x32)x(32x16) f32 matmul: 8 chained V_WMMA_F32_16X16X4_F32.
// Vertical pass is a per-pixel 31-tap gather (VALU).
// Tile staging: TDM tensor_load_to_lds for interior tiles (TENSORcnt),
// global_load_async_to_lds_b32 + zero-fill for border tiles (ASYNCcnt);
// the next channel's DMA is issued under the current channel's vertical pass.
// ---------------------------------------------------------------------------

typedef __attribute__((ext_vector_type(2))) float v2f;
typedef __attribute__((ext_vector_type(8))) float v8f;
typedef __attribute__((ext_vector_type(4))) unsigned int u32x4;
typedef __attribute__((ext_vector_type(8))) unsigned int u32x8;

#define IMG_H   512
#define IMG_W   512
#define NCH     3
#define TILE    16
#define HALO    15
#define IN_H    46          // TILE + 2*HALO
#define IN_W    47          // cols 0..46: x_local + tap, x in [0,15], tap in [0,31]
#define TAPS    32          // 31 real taps centered at 15, padded to 32 (tap 31 == 0)
#define NPLANES 32
#define PHALF   16          // planes per WMMA phase (N dimension)

// Async global -> LDS copy, one dword per enabled lane (tracked by ASYNCcnt).
__device__ __forceinline__ void async_g2l_b32(uint32_t lds_byte_off, const float* gptr) {
    asm volatile("global_load_async_to_lds_b32 %0, %1, off"
                 :: "v"(lds_byte_off), "v"(gptr)
                 : "memory");
}

// TDM: one 2D tile (IN_H rows x IN_W cols of f32, row stride IMG_W) -> LDS.
// D# group layouts per CDNA5 ISA 8.3/8.4. Rows land contiguously in LDS
// (Laddr increments linearly), matching sIn's row-major layout.
__device__ __forceinline__ void tdm_load_tile(const float* gbase, uint32_t lds_off) {
    const uint64_t ga = (uint64_t)(uintptr_t)gbase;
    u32x4 g0;
    g0.x = 0x1u;                                   // count=1, user mode
    g0.y = lds_off;                                // lds_addr (bytes)
    g0.z = (uint32_t)ga;                           // global_addr[31:0]
    g0.w = ((uint32_t)(ga >> 32) & 0x01FFFFFFu)    // global_addr[56:32]
         | 0x80000000u;                            // type=2 ("image") bits 127:126
    u32x8 g1;
    g1[0] = 0x00020000u;                           // data_size=2 (4 bytes); no multicast
    g1[1] = ((uint32_t)IMG_W & 0xFFFFu) << 16;     // tensor_dim0[15:0]
    g1[2] = ((uint32_t)IMG_H & 0xFFFFu) << 16;     // tensor_dim0[31:16]=0 | tensor_dim1[15:0]
    g1[3] = (uint32_t)IN_W << 16;                  // tensor_dim1[31:16]=0 | tile_dim0
    g1[4] = (uint32_t)IN_H;                        // tile_dim1 (tile_dim2=0)
    g1[5] = (uint32_t)IMG_W;                       // tensor_dim0_stride[31:0]
    g1[6] = 0u;                                    // stride[47:32] | dim1_stride[15:0]
    g1[7] = 0u;
    asm volatile("tensor_load_to_lds %0, %1"
                 :: "s"(g0), "s"(g1)
                 : "memory");
}

// Issue the haloed tile load for one channel.
__device__ __forceinline__ void load_tile(const float* chimg, float* sIn,
                                          int x0, int y0, bool interior, int tid) {
    if (interior) {
        // Scalar-branch gate: TDM ignores EXEC, so only wave 0 may reach it.
        if (__builtin_amdgcn_readfirstlane(tid) < 32) {
            tdm_load_tile(chimg + (size_t)(y0 - HALO) * IMG_W + (x0 - HALO),
                          (uint32_t)(uintptr_t)sIn);
        }
    } else {
        for (int e = tid; e < IN_H * IN_W; e += 256) {
            const int row = e / IN_W;
            const int col = e - row * IN_W;
            const int gy  = y0 - HALO + row;
            const int gx  = x0 - HALO + col;
            if ((unsigned)gy < (unsigned)IMG_H && (unsigned)gx < (unsigned)IMG_W) {
                async_g2l_b32((uint32_t)(uintptr_t)(&sIn[e]),
                              chimg + (size_t)gy * IMG_W + gx);
            } else {
                sIn[e] = 0.0f;                     // zero 'same' padding
            }
        }
    }
}

__device__ __forceinline__ void wait_tile_loads() {
    asm volatile("s_wait_asynccnt 0" ::: "memory");
    __builtin_amdgcn_s_wait_tensorcnt(0);
}

// Stage 1: per image row r, H[x, plane] = sum_t In[r, x+t] * g_plane[t]
//   A (16x4 f32): lanes 0-15 hold K={0,1}, lanes 16-31 K={2,3} (ISA 7.12.2)
//   B (4x16 f32): mirrored K split, col = lane&15
//   D (16x16 f32): VGPR i -> M = i + 8*(lane>>4), N = lane&15
__device__ __forceinline__ void stage1_wmma(const float* sIn, float* sH,
                                            const float* sBank, int phase,
                                            int wave, int m, int hi, int kh) {
    for (int r = wave; r < IN_H; r += 8) {
        v8f d = {};
        const float* inRow = &sIn[r * IN_W];
        const float* bank  = &sBank[(phase * PHALF + m) * TAPS];
#pragma unroll
        for (int kc = 0; kc < 8; ++kc) {
            const int t0 = kc * 4 + kh;
            v2f a, bb;
            a.x  = inRow[m + t0];
            a.y  = inRow[m + t0 + 1];
            bb.x = bank[t0];
            bb.y = bank[t0 + 1];
            d = __builtin_amdgcn_wmma_f32_16x16x4_f32(
                    false, a, false, bb, (short)0, d, false, false);
        }
#pragma unroll
        for (int i = 0; i < 8; ++i)
            sH[(r * TILE + (i + 8 * hi)) * PHALF + m] = d[i];
    }
}

// Stage 2: vertical 31-tap pass for pixels whose plane lives in this phase.
__device__ __forceinline__ void stage2_vert(const float* sH, const float* sBank,
                                            const int* sPlane, int phase,
                                            int tid, float* acc) {
    const int ly = tid >> 4, lx = tid & 15;
    const int p  = sPlane[tid];
    if ((p >> 4) == phase) {
        const int    pl = p & 15;
        const float* g  = &sBank[p * TAPS];
        float a = 0.0f;
        for (int dy = 0; dy < 31; ++dy)
            a += g[dy] * sH[((ly + dy) * TILE + lx) * PHALF + pl];
        *acc = a;
    }
}

__launch_bounds__(256)
__global__ void defocus_tile_kernel(const float* __restrict__ img,
                                    const float* __restrict__ coc,
                                    float* __restrict__ out) {
    __shared__ float sIn[IN_H * IN_W];            //  8648 B  input tile + halo (1 ch)
    __shared__ float sH[IN_H * TILE * PHALF];     // 47104 B  H[row][x][plane%16]
    __shared__ float sBank[NPLANES * TAPS];       //  4096 B  padded 1-D Gaussians
    __shared__ int   sPlane[TILE * TILE];         //  1024 B  per-pixel plane id

    const int tid  = threadIdx.x;
    const int lane = tid & 31;
    const int wave = tid >> 5;
    const int m    = lane & 15;
    const int hi   = lane >> 4;
    const int kh   = hi * 2;

    const int x0 = blockIdx.x * TILE;
    const int y0 = blockIdx.y * TILE;
    const int b  = blockIdx.z;
    const bool interior = (x0 >= HALO) && (x0 + TILE + HALO + 1 <= IMG_W) &&
                          (y0 >= HALO) && (y0 + TILE + HALO     <= IMG_H);

    // Kick off channel 0's tile DMA first so it overlaps bank construction.
    load_tile(img + (size_t)(b * NCH + 0) * IMG_H * IMG_W, sIn, x0, y0, interior, tid);

    // ---- Padded Gaussian bank (matches the numpy construction) ----
    for (int e = tid; e < NPLANES * TAPS; e += 256) {
        const int   pi = e >> 5;
        const int   t  = e & 31;
        const float cv = (float)pi * (50.0f / 31.0f);
        float val = 0.0f;
        if (t < 31) {
            if (cv < 0.5f) {
                val = (t == HALO) ? 1.0f : 0.0f;           // identity plane
            } else {
                int k = (int)(2.0f * cv + 1.0f);
                if ((k & 1) == 0) k += 1;
                if (k > 31) k = 31;
                const int r = k >> 1;
                const int j = t - HALO;
                if (j >= -r && j <= r) {
                    const float sig    = cv / 2.355f;
                    const float inv2s2 = 1.0f / (2.0f * sig * sig);
                    float sum = 0.0f;
                    for (int u = -r; u <= r; ++u)
                        sum += expf(-(float)(u * u) * inv2s2);
                    val = expf(-(float)(j * j) * inv2s2) / sum;
                }
            }
        }
        sBank[e] = val;
    }

    // ---- Per-pixel plane bucket (hard select, mirrors the weight masks) ----
    {
        const int ly = tid >> 4, lx = tid & 15;
        const float c = coc[((size_t)b * IMG_H + (y0 + ly)) * IMG_W + (x0 + lx)];
        int p = (int)floorf(c * (31.0f / 50.0f) + 0.5f);
        p = p < 0 ? 0 : (p > 31 ? 31 : p);
        sPlane[tid] = p;
    }

    wait_tile_loads();
    __syncthreads();

    for (int ch = 0; ch < NCH; ++ch) {
        float acc = 0.0f;

        // ---- Phase 0 (planes 0..15) ----
        stage1_wmma(sIn, sH, sBank, 0, wave, m, hi, kh);
        __syncthreads();
        stage2_vert(sH, sBank, sPlane, 0, tid, &acc);
        __syncthreads();                           // sH reuse by phase 1

        // ---- Phase 1 (planes 16..31) ----
        stage1_wmma(sIn, sH, sBank, 1, wave, m, hi, kh);
        __syncthreads();                           // last read of sIn this channel

        // Prefetch next channel's tile under the vertical pass + store.
        if (ch + 1 < NCH)
            load_tile(img + (size_t)(b * NCH + ch + 1) * IMG_H * IMG_W,
                      sIn, x0, y0, interior, tid);

        stage2_vert(sH, sBank, sPlane, 1, tid, &acc);

        const int ly = tid >> 4, lx = tid & 15;
        out[((size_t)(b * NCH + ch) * IMG_H + (y0 + ly)) * IMG_W + (x0 + lx)] = acc;

        wait_tile_loads();                         // next tile resident in sIn
        __syncthreads();
    }
}

// ---------------------------------------------------------------------------
// Harness entry point.
// d_in[0] = sharp_image f32 [4,3,512,512], d_in[1] = coc_map f32 [4,1,512,512]
// d_out   = f32 [4,3,512,512]
// ---------------------------------------------------------------------------
extern "C" void kernel_launch(void* const* d_in, const int* in_sizes, int n_in,
                              void* d_out, int out_size, void* d_ws, size_t ws_size,
                              hipStream_t stream) {
    (void)in_sizes; (void)n_in; (void)out_size; (void)d_ws; (void)ws_size;
    const float* img = (const float*)d_in[0];
    const float* coc = (const float*)d_in[1];
    float* out = (float*)d_out;

    dim3 grid(IMG_W / TILE, IMG_H / TILE, 4);   // 32 x 32 x 4 tiles
    dim3 block(256);                            // 8 wave32 per workgroup
    defocus_tile_kernel<<<grid, block, 0, stream>>>(img, coc, out);
}